// GCN_74354473828951
// MI455X (gfx1250) — compile-verified
//
#include <hip/hip_runtime.h>
#include <math.h>

typedef __attribute__((ext_vector_type(2))) float v2f;
typedef __attribute__((ext_vector_type(8))) float v8f;

#define C_DIM 64

// ---------------- degree / normalization ----------------

__global__ void gcn_deg_init(float* __restrict__ deg, int n) {
    int i = blockIdx.x * blockDim.x + threadIdx.x;
    if (i < n) deg[i] = 1.0f;   // self-loop contribution
}

__global__ void gcn_deg_accum(const int* __restrict__ dst, float* __restrict__ deg, int E) {
    int i = blockIdx.x * blockDim.x + threadIdx.x;
    if (i < E) atomicAdd(&deg[dst[i]], 1.0f);
}

__global__ void gcn_deg_rsqrt(float* __restrict__ deg, int n) {
    int i = blockIdx.x * blockDim.x + threadIdx.x;
    if (i < n) deg[i] = rsqrtf(deg[i]);   // deg >= 1 always
}

// ---------------- GEMM: out[M x 64] = (relu?)A[M x 64] @ W[64 x 64] ----------------
// One wave computes a 16x64 strip via 4 accumulators of V_WMMA_F32_16X16X4_F32.
// Requires M % 16 == 0 (true here: 100000).

__global__ void gcn_gemm_wmma(const float* __restrict__ A,
                              const float* __restrict__ W,
                              float* __restrict__ out,
                              int M, int relu_a) {
    const int wave = threadIdx.x >> 5;
    const int lane = threadIdx.x & 31;
    const int m0 = (blockIdx.x * (blockDim.x >> 5) + wave) << 4;
    if (m0 >= M) return;              // wave-uniform: EXEC stays all-1s for WMMA

    const int lm = lane & 15;         // col-in-tile / row-in-tile index
    const int kh = lane >> 4;         // K-half select (0 or 1)

    v8f acc[4];
#pragma unroll
    for (int nt = 0; nt < 4; ++nt)
#pragma unroll
        for (int r = 0; r < 8; ++r) acc[nt][r] = 0.0f;

    const float* arow = A + (size_t)(m0 + lm) * C_DIM;

    for (int kk = 0; kk < C_DIM; kk += 4) {
        const int k0 = kk + (kh << 1);
        v2f a;
        a[0] = arow[k0];
        a[1] = arow[k0 + 1];
        if (relu_a) {
            a[0] = fmaxf(a[0], 0.0f);
            a[1] = fmaxf(a[1], 0.0f);
        }
#pragma unroll
        for (int nt = 0; nt < 4; ++nt) {
            v2f b;
            b[0] = W[(size_t)k0 * C_DIM + nt * 16 + lm];
            b[1] = W[(size_t)(k0 + 1) * C_DIM + nt * 16 + lm];
            acc[nt] = __builtin_amdgcn_wmma_f32_16x16x4_f32(
                false, a, false, b, (short)0, acc[nt], false, false);
        }
    }

    // C/D layout: VGPR r -> row (r + 8*kh), lanes%16 -> col within 16-wide tile
#pragma unroll
    for (int nt = 0; nt < 4; ++nt)
#pragma unroll
        for (int r = 0; r < 8; ++r)
            out[(size_t)(m0 + r + (kh << 3)) * C_DIM + nt * 16 + lm] = acc[nt][r];
}

// ---------------- aggregation ----------------
// out[v][c] = bias[c] + h[v][c] * dinv[v]^2   (self-loop term; also full init)

__global__ void gcn_agg_init(const float* __restrict__ h,
                             const float* __restrict__ dinv,
                             const float* __restrict__ bias,
                             float* __restrict__ out, int total) {
    int i = blockIdx.x * blockDim.x + threadIdx.x;
    if (i < total) {
        int v = i >> 6;
        int c = i & 63;
        float dv = dinv[v];
        out[i] = bias[c] + h[i] * dv * dv;
    }
}

// one wave per edge; each lane handles 2 channels; fp32 scatter atomics hit L2
__global__ void gcn_agg_edges(const int* __restrict__ src,
                              const int* __restrict__ dst,
                              const float* __restrict__ h,
                              const float* __restrict__ dinv,
                              float* __restrict__ out, int E) {
    int g = blockIdx.x * blockDim.x + threadIdx.x;
    int e = g >> 5;
    if (e >= E) return;
    int lane = g & 31;
    int s = src[e];
    int d = dst[e];
    float norm = dinv[s] * dinv[d];
    int c = lane << 1;
    const float2 hv = *(const float2*)(h + (size_t)s * C_DIM + c);
    float* o = out + (size_t)d * C_DIM + c;
    atomicAdd(o,     hv.x * norm);
    atomicAdd(o + 1, hv.y * norm);
}

// ---------------- launch ----------------

extern "C" void kernel_launch(void* const* d_in, const int* in_sizes, int n_in,
                              void* d_out, int out_size, void* d_ws, size_t ws_size,
                              hipStream_t stream) {
    const float* x  = (const float*)d_in[0];
    const int*   ei = (const int*)  d_in[1];
    const float* W1 = (const float*)d_in[2];
    const float* b1 = (const float*)d_in[3];
    const float* W2 = (const float*)d_in[4];
    const float* b2 = (const float*)d_in[5];

    const int n = in_sizes[0] / C_DIM;   // 100000
    const int E = in_sizes[1] / 2;       // 1600000
    const int* src = ei;
    const int* dst = ei + E;
    float* out = (float*)d_out;

    // workspace layout: dinv[n] | h[n*64] | agg1[n*64]   (~51.6 MB)
    char* ws = (char*)d_ws;
    float* dinv = (float*)ws;
    size_t off = (((size_t)n * sizeof(float)) + 255) & ~(size_t)255;
    float* h = (float*)(ws + off);
    off += (((size_t)n * C_DIM * sizeof(float)) + 255) & ~(size_t)255;
    float* agg1 = (float*)(ws + off);

    const int tb = 256;
    const int nodeGrid = (n + tb - 1) / tb;
    const int edgeGrid = (E + tb - 1) / tb;
    const int featGrid = ((n * C_DIM) + tb - 1) / tb;
    const int edgeWaveGrid = (int)(((long long)E * 32 + tb - 1) / tb);
    const int wavesPerBlock = tb / 32;
    const int mtiles = n / 16;           // n % 16 == 0
    const int gemmGrid = (mtiles + wavesPerBlock - 1) / wavesPerBlock;

    // normalization: deg -> rsqrt(deg) in place
    gcn_deg_init  <<<nodeGrid, tb, 0, stream>>>(dinv, n);
    gcn_deg_accum <<<edgeGrid, tb, 0, stream>>>(dst, dinv, E);
    gcn_deg_rsqrt <<<nodeGrid, tb, 0, stream>>>(dinv, n);

    // layer 1: h = x @ W1; agg1 = b1 + selfloop + edge scatter
    gcn_gemm_wmma <<<gemmGrid, tb, 0, stream>>>(x, W1, h, n, 0);
    gcn_agg_init  <<<featGrid, tb, 0, stream>>>(h, dinv, b1, agg1, n * C_DIM);
    gcn_agg_edges <<<edgeWaveGrid, tb, 0, stream>>>(src, dst, h, dinv, agg1, E);

    // layer 2: h = relu(agg1) @ W2 (relu fused into A load); out = b2 + selfloop + scatter
    gcn_gemm_wmma <<<gemmGrid, tb, 0, stream>>>(agg1, W2, h, n, 1);
    gcn_agg_init  <<<featGrid, tb, 0, stream>>>(h, dinv, b2, out, n * C_DIM);
    gcn_agg_edges <<<edgeWaveGrid, tb, 0, stream>>>(src, dst, h, dinv, out, E);
}